// SearchTopK_InnerProduct_34514357191021
// MI455X (gfx1250) — compile-verified
//
#include <hip/hip_runtime.h>

typedef __attribute__((ext_vector_type(2))) float v2f;
typedef __attribute__((ext_vector_type(8))) float v8f;

#define NQ_TOTAL 256
#define NDIM     128
#define NCORP    1000000
#define TOPK     100
#define CHUNK    2048
#define NCHUNKS  ((NCORP + CHUNK - 1) / CHUNK)   /* 489 */
#define MTILES   (NQ_TOTAL / 16)                 /* 16  */
#define WAVES    8
#define BLOCK    256

#define SG_WMMA  0x008  /* MFMA/WMMA scheduling group */
#define SG_VMRD  0x020  /* VMEM read scheduling group */

__device__ __forceinline__ unsigned ordkey(float f) {
  // monotone map float -> unsigned (larger float => larger key), -inf safe
  unsigned u = __float_as_uint(f);
  return u ^ (((unsigned)((int)u >> 31)) | 0x80000000u);
}

// ---------------------------------------------------------------------------
// Kernel 1: WMMA fp32 GEMM tile (16 queries x 2048 corpus cols) -> LDS scores,
// then exact per-query top-100 of the chunk via 4-level radix-256 descend.
// ---------------------------------------------------------------------------
__global__ void __launch_bounds__(BLOCK)
gemm_chunk_topk(const float* __restrict__ Q, const float* __restrict__ C,
                float* __restrict__ candS, int* __restrict__ candI) {
  extern __shared__ unsigned char smem_raw[];
  float*    sc      = (float*)smem_raw;                           // [16][CHUNK]
  unsigned* histAll = (unsigned*)(smem_raw + 16 * CHUNK * 4);     // [8][256]
  unsigned* cntG    = histAll + WAVES * 256;                      // [8]
  unsigned* cntE    = cntG + WAVES;                               // [8]

  const int lane  = threadIdx.x & 31;
  const int wave  = threadIdx.x >> 5;
  const int mtile = blockIdx.x;
  const int chunk = blockIdx.y;
  const int lm    = lane & 15;   // M row (A) / N col (B,C)
  const int kh    = lane >> 4;   // K-half: VGPR0 holds K, K+2 ; VGPR1 K+1, K+3

  // ---- A operands: 16 queries x 128, 32 K-slices of v2f (resident in VGPRs)
  v2f A[32];
  {
    const float* qrow = Q + (size_t)(mtile * 16 + lm) * NDIM + 2 * kh;
#pragma unroll
    for (int s = 0; s < 32; ++s) A[s] = *(const v2f*)(qrow + 4 * s);
  }

  const int colbase = chunk * CHUNK;
  const float NEG = -__builtin_inff();

  // ---- GEMM: each wave owns 256 columns; two C-tiles (32 cols) per step ----
  for (int c0 = wave * 256; c0 < wave * 256 + 256; c0 += 32) {
    int ca = colbase + c0 + lm;
    int cb = ca + 16;
    const float* rA = C + (size_t)(ca < NCORP ? ca : NCORP - 1) * NDIM + 2 * kh;
    const float* rB = C + (size_t)(cb < NCORP ? cb : NCORP - 1) * NDIM + 2 * kh;
    v8f accA = {};
    v8f accB = {};

    // Double-buffered B operands: 8 K-slices (16 b64 loads) per batch.
    v2f bA[2][8], bB[2][8];
#pragma unroll
    for (int s = 0; s < 8; ++s) {
      bA[0][s] = *(const v2f*)(rA + 4 * s);
      bB[0][s] = *(const v2f*)(rB + 4 * s);
    }
#pragma unroll
    for (int g = 0; g < 4; ++g) {
      const int cur = g & 1;
      const int nxt = cur ^ 1;
      if (g < 3) {
#pragma unroll
        for (int s = 0; s < 8; ++s) {
          bA[nxt][s] = *(const v2f*)(rA + 4 * (8 * (g + 1) + s));
          bB[nxt][s] = *(const v2f*)(rB + 4 * (8 * (g + 1) + s));
        }
      }
#pragma unroll
      for (int s = 0; s < 8; ++s) {
        accA = __builtin_amdgcn_wmma_f32_16x16x4_f32(false, A[8 * g + s], false,
                                                     bA[cur][s], (short)0, accA,
                                                     false, false);
        accB = __builtin_amdgcn_wmma_f32_16x16x4_f32(false, A[8 * g + s], false,
                                                     bB[cur][s], (short)0, accB,
                                                     false, false);
      }
    }
    // Pin schedule: loads stay >=1 batch (16 b64) ahead of the WMMA groups,
    // and the two accumulator chains interleave inside each WMMA group.
    __builtin_amdgcn_sched_group_barrier(SG_VMRD, 32, 0); // L0 + L1
    __builtin_amdgcn_sched_group_barrier(SG_WMMA, 16, 0); // W0
    __builtin_amdgcn_sched_group_barrier(SG_VMRD, 16, 0); // L2
    __builtin_amdgcn_sched_group_barrier(SG_WMMA, 16, 0); // W1
    __builtin_amdgcn_sched_group_barrier(SG_VMRD, 16, 0); // L3
    __builtin_amdgcn_sched_group_barrier(SG_WMMA, 32, 0); // W2 + W3

    // C layout: VGPR j -> M = j (lanes 0-15) / j+8 (lanes 16-31), N = lane%16
#pragma unroll
    for (int j = 0; j < 8; ++j) {
      int qr = j + 8 * kh;
      sc[qr * CHUNK + c0 + lm]      = (ca < NCORP) ? accA[j] : NEG;
      sc[qr * CHUNK + c0 + 16 + lm] = (cb < NCORP) ? accB[j] : NEG;
    }
  }
  __syncthreads();

  // ---- per-query exact top-100 in chunk; wave-private histograms ----
  unsigned* hw = histAll + wave * 256;
  for (int qq = 0; qq < 2; ++qq) {
    int q = wave + qq * 8;
    const float* row = sc + q * CHUNK;
    unsigned pref = 0;
    int need = TOPK;
    for (int shift = 24; shift >= 0; shift -= 8) {
#pragma unroll
      for (int b = lane; b < 256; b += 32) hw[b] = 0;
      for (int i = lane; i < CHUNK; i += 32) {
        unsigned k = ordkey(row[i]);
        if ((((unsigned long long)(k ^ pref)) >> (shift + 8)) == 0ull)
          atomicAdd(&hw[(k >> shift) & 255], 1u);
      }
      unsigned sel = 0;
      int nd = need;
      if (lane == 0) {
        for (int b = 255; b >= 0; --b) {
          int c = (int)hw[b];
          if (c >= nd) { sel = (unsigned)b; break; }
          nd -= c;
        }
      }
      sel  = (unsigned)__shfl((int)sel, 0);
      nd   = __shfl(nd, 0);
      pref |= sel << shift;
      need  = nd;
    }
    const unsigned T = pref;        // exact key of 100th-largest in chunk
    if (lane == 0) { cntG[wave] = 0; cntE[wave] = 0; }  // wave DS is in-order
    const int nG = TOPK - need;     // exact count of keys strictly > T
    size_t baseOut = ((size_t)(mtile * 16 + q) * NCHUNKS + chunk) * TOPK;
    for (int i = lane; i < CHUNK; i += 32) {
      float s = row[i];
      unsigned k = ordkey(s);
      int pos = -1;
      if (k > T) {
        pos = (int)atomicAdd(&cntG[wave], 1u);
      } else if (k == T) {
        int e = (int)atomicAdd(&cntE[wave], 1u);
        if (e < need) pos = nG + e;
      }
      if (pos >= 0 && pos < TOPK) {
        candS[baseOut + pos] = s;
        candI[baseOut + pos] = colbase + i;
      }
    }
  }
}

// ---------------------------------------------------------------------------
// Kernel 2: per query, merge 489*100 candidates -> exact top-100, rank-sort,
// emit ids (as float), sorted scores, and gathered embeddings.
// ---------------------------------------------------------------------------
__global__ void __launch_bounds__(BLOCK)
merge_topk(const float* __restrict__ C, const int* __restrict__ ids,
           const float* __restrict__ candS, const int* __restrict__ candI,
           float* __restrict__ out) {
  const int q   = blockIdx.x;
  const int tid = threadIdx.x;
  const int ncand = NCHUNKS * TOPK;

  __shared__ unsigned hist[256];
  __shared__ float    tops[TOPK];
  __shared__ int      topi[TOPK];
  __shared__ int      sidx[TOPK];
  __shared__ unsigned cg, ce;
  __shared__ unsigned s_sel;
  __shared__ int      s_need;

  const float* cs = candS + (size_t)q * ncand;
  const int*   ci = candI + (size_t)q * ncand;

  unsigned pref = 0;
  int need = TOPK;
  for (int shift = 24; shift >= 0; shift -= 8) {
    if (tid < 256) hist[tid] = 0;
    __syncthreads();
    for (int i = tid; i < ncand; i += BLOCK) {
      unsigned k = ordkey(cs[i]);
      if ((((unsigned long long)(k ^ pref)) >> (shift + 8)) == 0ull)
        atomicAdd(&hist[(k >> shift) & 255], 1u);
    }
    __syncthreads();
    if (tid == 0) {
      unsigned sel = 0;
      int nd = need;
      for (int b = 255; b >= 0; --b) {
        int c = (int)hist[b];
        if (c >= nd) { sel = (unsigned)b; break; }
        nd -= c;
      }
      s_sel = sel;
      s_need = nd;
    }
    __syncthreads();
    pref |= s_sel << shift;
    need  = s_need;
    __syncthreads();
  }
  const unsigned T = pref;
  if (tid == 0) { cg = 0; ce = 0; }
  __syncthreads();
  const int nG = TOPK - need;
  for (int i = tid; i < ncand; i += BLOCK) {
    float s = cs[i];
    unsigned k = ordkey(s);
    int pos = -1;
    if (k > T) {
      pos = (int)atomicAdd(&cg, 1u);
    } else if (k == T) {
      int e = (int)atomicAdd(&ce, 1u);
      if (e < need) pos = nG + e;
    }
    if (pos >= 0 && pos < TOPK) { tops[pos] = s; topi[pos] = ci[i]; }
  }
  __syncthreads();

  float* outIds    = out;                              // [256][100]
  float* outScores = out + (size_t)NQ_TOTAL * TOPK;    // [256][100]
  float* outEmb    = out + (size_t)2 * NQ_TOTAL * TOPK;// [256][100][128]

  if (tid < TOPK) {
    float ms = tops[tid];
    int   mi = topi[tid];
    int rank = 0;
    for (int j = 0; j < TOPK; ++j) {
      float s2 = tops[j];
      int   i2 = topi[j];
      if (s2 > ms || (s2 == ms && i2 < mi)) ++rank;
    }
    int safe = (mi < 0) ? 0 : (mi >= NCORP ? NCORP - 1 : mi);
    outIds[(size_t)q * TOPK + rank]    = (float)ids[safe];
    outScores[(size_t)q * TOPK + rank] = ms;
    sidx[rank] = safe;
  }
  __syncthreads();
  for (int e = tid; e < TOPK * NDIM; e += BLOCK) {
    int r = e >> 7, c = e & (NDIM - 1);
    outEmb[((size_t)q * TOPK + r) * NDIM + c] = C[(size_t)sidx[r] * NDIM + c];
  }
}

// ---------------------------------------------------------------------------
extern "C" void kernel_launch(void* const* d_in, const int* in_sizes, int n_in,
                              void* d_out, int out_size, void* d_ws, size_t ws_size,
                              hipStream_t stream) {
  (void)in_sizes; (void)n_in; (void)out_size; (void)ws_size;
  const float* Q   = (const float*)d_in[0];   // [256][128]
  const float* C   = (const float*)d_in[1];   // [1000000][128]
  const int*   ids = (const int*)d_in[2];     // [1000000]
  float* out = (float*)d_out;

  size_t candElems = (size_t)NQ_TOTAL * NCHUNKS * TOPK;
  float* candS = (float*)d_ws;
  int*   candI = (int*)((unsigned char*)d_ws + candElems * sizeof(float));

  size_t smem = (size_t)16 * CHUNK * 4 + (size_t)WAVES * 256 * 4 + 64;
  hipFuncSetAttribute(reinterpret_cast<const void*>(gemm_chunk_topk),
                      hipFuncAttributeMaxDynamicSharedMemorySize, (int)smem);

  gemm_chunk_topk<<<dim3(MTILES, NCHUNKS), BLOCK, smem, stream>>>(Q, C, candS, candI);
  merge_topk<<<NQ_TOTAL, BLOCK, 0, stream>>>(C, ids, candS, candI, out);
}